// RNNBlock_2078764171552
// MI455X (gfx1250) — compile-verified
//
#include <hip/hip_runtime.h>
#include <hip/hip_bf16.h>

// B=64, T=512, I=256, H=512, fp32 2-layer tanh RNN.
// proj: xw = inp @ Wih^T   (big parallel WMMA f32 GEMM)
// recur: h_t = tanh(xw_t + h_{t-1} @ Whh^T), persistent kernel, one WG per
//        16-row batch slice (no cross-WG deps), h double-buffered in LDS.

typedef float v2f __attribute__((ext_vector_type(2)));
typedef float v8f __attribute__((ext_vector_type(8)));

#define RNN_B 64
#define RNN_T 512
#define RNN_H 512
#define LDS_STRIDE 516   // 516 % 64 == 4 -> conflict-free ds_load_b64 A-frags

// gfx1250 has a native V_TANH_F32 transcendental; use it if clang exposes it.
__device__ __forceinline__ float fast_tanh(float x) {
#if __has_builtin(__builtin_amdgcn_tanhf)
  return __builtin_amdgcn_tanhf(x);
#elif __has_builtin(__builtin_amdgcn_tanh_f32)
  return __builtin_amdgcn_tanh_f32(x);
#else
  return tanhf(x);
#endif
}

// ---------------------------------------------------------------------------
// Input projection: C[M,N] = A[M,K] * W[N,K]^T, M = B*T = 32768, N = H = 512.
// Each wave owns 4 M-tiles x 1 N-tile (B-fragment reused 4x).
// grid = (M/64, N/(16*8)), block = 256 (8 waves).
// ---------------------------------------------------------------------------
template <int K>
__global__ __launch_bounds__(256) void proj_kernel(const float* __restrict__ A,
                                                   const float* __restrict__ W,
                                                   float* __restrict__ C) {
  const int N    = RNN_H;
  const int wave = threadIdx.x >> 5;
  const int lane = threadIdx.x & 31;
  const int half = lane >> 4;   // 0: K-pair {0,1}, 1: K-pair {2,3}
  const int lr   = lane & 15;

  const int m0 = blockIdx.x * 64;                    // 4 M-tiles of 16
  const int n0 = (blockIdx.y * 8 + wave) * 16;       // 1 N-tile of 16

  const float* __restrict__ arow = A + (size_t)(m0 + lr) * K + half * 2;
  const float* __restrict__ brow = W + (size_t)(n0 + lr) * K + half * 2;

  v8f acc[4] = {};
#pragma unroll 4
  for (int k = 0; k < K; k += 4) {
    v2f b = *(const v2f*)(brow + k);
#pragma unroll
    for (int i = 0; i < 4; ++i) {
      v2f a = *(const v2f*)(arow + (size_t)i * 16 * K + k);
      acc[i] = __builtin_amdgcn_wmma_f32_16x16x4_f32(
          false, a, false, b, (short)0, acc[i], false, false);
    }
  }

#pragma unroll
  for (int i = 0; i < 4; ++i) {
#pragma unroll
    for (int r = 0; r < 8; ++r) {
      const int m = m0 + i * 16 + r + half * 8;
      C[(size_t)m * N + n0 + lr] = acc[i][r];
    }
  }
}

// ---------------------------------------------------------------------------
// Recurrence for one layer. grid = 4 (one WG per 16 batch rows),
// block = 1024 (32 waves; wave w owns output columns [16w, 16w+16)).
// h kept in LDS, double-buffered; Whh stays in L2 (1 MB).
// xw fragment for step t+1 is prefetched into registers during step t's
// K-loop so its global-load latency hides behind the 128 WMMAs.
// ---------------------------------------------------------------------------
__global__ __launch_bounds__(1024) void recur_kernel(const float* __restrict__ xw,
                                                     const float* __restrict__ Whh,
                                                     float* __restrict__ out) {
  __shared__ float hsh[2][16 * LDS_STRIDE];   // 2 x 33 KB

  const int wave = threadIdx.x >> 5;
  const int lane = threadIdx.x & 31;
  const int half = lane >> 4;
  const int lr   = lane & 15;
  const int n0   = wave * 16;
  const int b0   = blockIdx.x * 16;

  // h0 = 0
  for (int i = threadIdx.x; i < 16 * LDS_STRIDE; i += 1024) hsh[0][i] = 0.0f;
  __syncthreads();

  const float* __restrict__ brow =
      Whh + (size_t)(n0 + lr) * RNN_H + half * 2;   // B-frag base (L2 resident)

  const size_t row_stride = (size_t)RNN_T * RNN_H;  // stride between batch rows

  // Preload xw for t = 0.
  v8f xw_reg;
#pragma unroll
  for (int r = 0; r < 8; ++r) {
    const int m = r + half * 8;
    xw_reg[r] = xw[(size_t)(b0 + m) * row_stride + n0 + lr];
  }

  for (int t = 0; t < RNN_T; ++t) {
    const float* hcur = hsh[t & 1];
    float* hnext      = hsh[(t + 1) & 1];

    v8f c = xw_reg;

    // Prefetch next step's xw (clamped index keeps the last step in bounds;
    // its value is never consumed).
    const int tn = (t + 1 < RNN_T) ? t + 1 : t;
#pragma unroll
    for (int r = 0; r < 8; ++r) {
      const int m = r + half * 8;
      xw_reg[r] =
          xw[(size_t)(b0 + m) * row_stride + (size_t)tn * RNN_H + n0 + lr];
    }

    // c += h_{t-1}[16,512] @ Whh^T[:, n0:n0+16]  (K = 512, f32 WMMA)
    const float* arow = hcur + lr * LDS_STRIDE + half * 2;
#pragma unroll 8
    for (int k = 0; k < RNN_H; k += 4) {
      v2f a = *(const v2f*)(arow + k);          // ds_load_2addr_b64, no conflicts
      v2f b = *(const v2f*)(brow + k);          // global_load_b64 from L2
      c = __builtin_amdgcn_wmma_f32_16x16x4_f32(
          false, a, false, b, (short)0, c, false, false);
    }

    // Native v_tanh_f32, publish h_t to the other LDS buffer and to global.
#pragma unroll
    for (int r = 0; r < 8; ++r) {
      const int m   = r + half * 8;
      const float v = fast_tanh(c[r]);
      hnext[m * LDS_STRIDE + n0 + lr] = v;
      out[(size_t)(b0 + m) * row_stride + (size_t)t * RNN_H + n0 + lr] = v;
    }
    __syncthreads();   // all waves done reading hcur / writing hnext
  }
}

// ---------------------------------------------------------------------------
extern "C" void kernel_launch(void* const* d_in, const int* in_sizes, int n_in,
                              void* d_out, int out_size, void* d_ws, size_t ws_size,
                              hipStream_t stream) {
  const float* x     = (const float*)d_in[0];  // [B,T,I]
  const float* W_ih0 = (const float*)d_in[1];  // [H,I]
  const float* W_hh0 = (const float*)d_in[2];  // [H,H]
  const float* W_ih1 = (const float*)d_in[3];  // [H,H]
  const float* W_hh1 = (const float*)d_in[4];  // [H,H]
  float* out = (float*)d_out;                  // [B,T,H]
  float* xw  = (float*)d_ws;                   // [B,T,H] scratch (64 MB)

  const int M = RNN_B * RNN_T;                          // 32768
  dim3 pgrid(M / 64, RNN_H / (16 * 8));                 // (512, 4)

  // Layer 0
  proj_kernel<256><<<pgrid, 256, 0, stream>>>(x, W_ih0, xw);
  recur_kernel<<<4, 1024, 0, stream>>>(xw, W_hh0, out); // layer-0 h -> d_out

  // Layer 1 (reads layer-0 output from d_out, overwrites xw, then d_out)
  proj_kernel<512><<<pgrid, 256, 0, stream>>>(out, W_ih1, xw);
  recur_kernel<<<4, 1024, 0, stream>>>(xw, W_hh1, out);
}